// RoI_43078521979075
// MI455X (gfx1250) — compile-verified
//
#include <hip/hip_runtime.h>

// Problem constants (from reference): x[4,256,512,512] f32, 32 ROIs, 7x7 out.
#define OUTP 7
#define BATCH 4
#define CH 256
#define HH 512
#define WW 512
#define N_ROIS 32
#define WAVES_PER_BLOCK 8

typedef __attribute__((ext_vector_type(2))) float v2f;
typedef __attribute__((ext_vector_type(8))) float v8f;

// One wave handles one (b, c, roi): builds S[7][64] (row-bin column sums) in
// LDS, then P(7x7) = S * Wc via V_WMMA_F32_16X16X4_F32 with Wc the 0/1
// column-bin indicator, finally scales by 1/area and stores.
__global__ __launch_bounds__(256) void roi_adaptive_pool_wmma(
    const float* __restrict__ x, const int* __restrict__ rois,
    float* __restrict__ out)
{
  __shared__ float sS[WAVES_PER_BLOCK][OUTP * 64];

  const int lane = threadIdx.x & 31;
  const int wave = threadIdx.x >> 5;

  // blockIdx.x = ((b * N_ROIS) + roi) * (CH/8) + cb
  const int cb  = blockIdx.x % (CH / WAVES_PER_BLOCK);
  const int rem = blockIdx.x / (CH / WAVES_PER_BLOCK);
  const int roi = rem % N_ROIS;
  const int b   = rem / N_ROIS;
  const int c   = cb * WAVES_PER_BLOCK + wave;

  const int rx1 = rois[roi * 4 + 0];
  const int ry1 = rois[roi * 4 + 1];
  const int rx2 = rois[roi * 4 + 2];
  const int ry2 = rois[roi * 4 + 3];
  const int bw = rx2 - rx1;   // 16..64
  const int bh = ry2 - ry1;   // 16..64

  const float* base =
      x + ((size_t)(b * CH + c) * HH + (size_t)ry1) * WW + rx1;

  // ---- Phase 1: row-bin column sums. Bins may overlap (floor/ceil), so we
  // iterate each bin's own row range. S[i][x] = 0 for x >= bw by construction.
  #pragma unroll
  for (int i = 0; i < OUTP; ++i) {
    const int hs = (i * bh) / OUTP;
    const int he = ((i + 1) * bh + OUTP - 1) / OUTP;   // ceil
    float f0 = 0.f, f1 = 0.f;
    // Prefetch first row of the next bin while we chew on this one.
    if (lane < bw)
      __builtin_prefetch(base + (size_t)he * WW + lane, 0, 1);
    for (int y = hs; y < he; ++y) {
      const float* row = base + (size_t)y * WW;
      if (lane < bw)      f0 += row[lane];
      if (lane + 32 < bw) f1 += row[lane + 32];
    }
    sS[wave][i * 64 + lane]      = f0;
    sS[wave][i * 64 + lane + 32] = f1;
  }
  __syncthreads();   // uniform: loop trip counts depend only on roi

  // ---- Phase 2: P = S * Wc via f32 WMMA (16x16x4), K-loop over columns.
  // A layout (f32 16x4): lane m = M row; half h: VGPR0 -> K=2h, VGPR1 -> K=2h+1.
  // B layout (f32 4x16): lane&15 = N col; half h: VGPR0 -> K=2h, VGPR1 -> K=2h+1.
  const int m  = lane & 15;
  const int h2 = lane >> 4;
  const int n  = m;                         // column bin index for B operand
  int wsn = 0, wen = 0;
  if (n < OUTP) {
    wsn = (n * bw) / OUTP;
    wen = ((n + 1) * bw + OUTP - 1) / OUTP;
  }

  const float* Srow = sS[wave] + (m < OUTP ? m : (OUTP - 1)) * 64;
  const float mMask = (m < OUTP) ? 1.f : 0.f;

  v8f acc = {0.f, 0.f, 0.f, 0.f, 0.f, 0.f, 0.f, 0.f};
  for (int kc = 0; kc < bw; kc += 4) {      // bw is wave-uniform; EXEC all 1s
    const int xa = kc + 2 * h2;             // xa, xa+1 <= 63
    v2f A, Bv;
    A.x  = Srow[xa]     * mMask;
    A.y  = Srow[xa + 1] * mMask;
    Bv.x = (xa     >= wsn && xa     < wen) ? 1.f : 0.f;
    Bv.y = (xa + 1 >= wsn && xa + 1 < wen) ? 1.f : 0.f;
    acc = __builtin_amdgcn_wmma_f32_16x16x4_f32(
        /*neg_a=*/false, A, /*neg_b=*/false, Bv,
        /*c_mod=*/(short)0, acc, /*reuse_a=*/false, /*reuse_b=*/false);
  }

  // ---- Phase 3: D layout: lanes 0..6 hold j, VGPRs 0..6 hold i (h==0 half).
  if (h2 == 0 && n < OUTP) {
    const int wlen = wen - wsn;
    const size_t obase =
        (((size_t)b * (N_ROIS * CH) + (size_t)roi * CH + (size_t)c)
         * OUTP) * OUTP + (size_t)n;
    #pragma unroll
    for (int r = 0; r < OUTP; ++r) {
      const int hs = (r * bh) / OUTP;
      const int he = ((r + 1) * bh + OUTP - 1) / OUTP;
      const float area = (float)((he - hs) * wlen);
      out[obase + (size_t)r * OUTP] = acc[r] / area;
    }
  }
}

extern "C" void kernel_launch(void* const* d_in, const int* in_sizes, int n_in,
                              void* d_out, int out_size, void* d_ws, size_t ws_size,
                              hipStream_t stream) {
  (void)in_sizes; (void)n_in; (void)d_ws; (void)ws_size; (void)out_size;
  const float* x    = (const float*)d_in[0];
  const int*   rois = (const int*)d_in[1];
  float*       out  = (float*)d_out;
  // 4096 blocks x 256 threads = 32768 waves = B * C * N_ROIS.
  dim3 grid(BATCH * N_ROIS * (CH / WAVES_PER_BLOCK));
  roi_adaptive_pool_wmma<<<grid, 256, 0, stream>>>(x, rois, out);
}